// prj_module_31207232373147
// MI455X (gfx1250) — compile-verified
//
#include <hip/hip_runtime.h>
#include <math.h>
#include <stdint.h>

#define NN 256   // image side
#define VV 128   // views
#define DD 256   // detector bins
#define SS 256   // samples per ray
#define BB 2     // batch

typedef __attribute__((ext_vector_type(2))) float v2f;
typedef __attribute__((ext_vector_type(8))) float v8f;

// Geometry constants (match reference: L = N*sqrt(2)/2, linspace(-L, L, 256))
#define GEO_L  181.01933598375618f
#define GEO_DT (2.0f * GEO_L / 255.0f)   // == ds integration step (D == S == 256)
#define GEO_CC 127.5f

// ---------------------------------------------------------------------------
// Kernel 1: adaptive step size via WMMA (feature_vec[B,7] @ w_vec[7,1] + b)
// One wave32. A = 16x4 f32 chunks (K=0..3, K=4..7 padded), B = 4x16 (col 0).
// ---------------------------------------------------------------------------
__global__ void cw_kernel(const float* __restrict__ wf,
                          const float* __restrict__ gamma,
                          const float* __restrict__ beta,
                          const float* __restrict__ fv,    // [B,7]
                          const float* __restrict__ wv,    // [7]
                          const float* __restrict__ bvec,  // [1]
                          float* __restrict__ cw) {        // [B]
  const int lane = threadIdx.x & 31;
  const int row  = lane & 15;
  const int hi   = lane >> 4;   // 0: K=0,1 / 1: K=2,3 (per 32-bit A 16x4 layout)

  auto F = [&](int r, int k) -> float {
    return (r < BB && k < 7) ? fv[r * 7 + k] : 0.0f;
  };

  v2f a0, a1, b0, b1;
  a0.x = F(row, hi ? 2 : 0);  a0.y = F(row, hi ? 3 : 1);
  a1.x = F(row, hi ? 6 : 4);  a1.y = F(row, hi ? 7 : 5);  // K=7 -> 0 pad
  b0.x = 0.0f; b0.y = 0.0f; b1.x = 0.0f; b1.y = 0.0f;
  // B 4x16 layout: VGPR0 lanes0-15 K=0 (N=lane), lanes16-31 K=2; VGPR1: K=1/K=3
  if (lane == 0)  { b0.x = wv[0]; b0.y = wv[1]; b1.x = wv[4]; b1.y = wv[5]; }
  if (lane == 16) { b0.x = wv[2]; b0.y = wv[3]; b1.x = wv[6]; b1.y = 0.0f; }

  v8f c = {};
  c = __builtin_amdgcn_wmma_f32_16x16x4_f32(false, a0, false, b0, (short)0, c,
                                            false, false);
  c = __builtin_amdgcn_wmma_f32_16x16x4_f32(false, a1, false, b1, (short)0, c,
                                            false, false);

  if (lane == 0) {
    // D(M=b, N=0) lives in VGPR b of lane 0 (C/D 16x16 f32 layout)
    float weight = fmaf(wf[0], gamma[0], beta[0]);
    for (int bb = 0; bb < BB; ++bb) {
      float maxv = c[bb] + bvec[0];
      float wsel = (weight <= maxv) ? weight : maxv;
      cw[bb] = fmaxf(wsel, 0.01f);
    }
  }
}

// ---------------------------------------------------------------------------
// Kernel 2: forward projection residual  res[b,v,d] = (A x)[b,v,d] - proj
// One block per (b, v); 1024 threads = 32 waves for gather-latency hiding.
// Thread (d, chunk) integrates 64 of the 256 samples of ray d; the 4 chunk
// partials are combined in a fixed (deterministic) order via LDS.
// Image gathers hit L2 (512 KB working set vs 192 MB L2).
// ---------------------------------------------------------------------------
__global__ void fwd_res_kernel(const float* __restrict__ img,
                               const float* __restrict__ proj,
                               float* __restrict__ res) {
  __shared__ float partial[4 * DD];

  const int bv    = blockIdx.x;          // 0 .. B*V-1
  const int b     = bv / VV;
  const int v     = bv % VV;
  const int tid   = threadIdx.x;         // 0 .. 1023
  const int d     = tid & 255;
  const int chunk = tid >> 8;            // 0 .. 3 (64 samples each)

  __builtin_prefetch(&proj[(size_t)bv * DD + d], 0, 0);  // global_prefetch_b8

  const float theta = (float)((double)v * (3.14159265358979323846 / (double)VV));
  float st, ct;
  sincosf(theta, &st, &ct);
  const float t = fmaf((float)d, GEO_DT, -GEO_L);
  const float* __restrict__ imgb = img + (size_t)b * NN * NN;

  float acc = 0.0f;
  const int k0 = chunk * (SS / 4);
  for (int k = k0; k < k0 + (SS / 4); ++k) {
    const float s  = fmaf((float)k, GEO_DT, -GEO_L);
    const float x  = t * ct - s * st + GEO_CC;
    const float y  = t * st + s * ct + GEO_CC;
    const float xf = floorf(x);
    const float yf = floorf(y);
    const int j0 = (int)xf;
    const int i0 = (int)yf;
    const float fx = x - xf, fy = y - yf;
    const float w00 = (1.0f - fy) * (1.0f - fx);
    const float w01 = (1.0f - fy) * fx;
    const float w10 = fy * (1.0f - fx);
    const float w11 = fy * fx;
    auto fetch = [&](int ii, int jj, float w) -> float {
      const bool inb = (ii >= 0) & (ii < NN) & (jj >= 0) & (jj < NN);
      const int ic = min(max(ii, 0), NN - 1);
      const int jc = min(max(jj, 0), NN - 1);
      return inb ? w * imgb[ic * NN + jc] : 0.0f;
    };
    acc += fetch(i0,     j0,     w00);
    acc += fetch(i0,     j0 + 1, w01);
    acc += fetch(i0 + 1, j0,     w10);
    acc += fetch(i0 + 1, j0 + 1, w11);
  }
  partial[chunk * DD + d] = acc;
  __syncthreads();

  if (tid < DD) {
    // fixed summation order -> bitwise deterministic across replays
    const float sum = ((partial[tid] + partial[DD + tid]) +
                       partial[2 * DD + tid]) + partial[3 * DD + tid];
    const size_t o = (size_t)bv * DD + tid;
    res[o] = sum * GEO_DT - proj[o];
  }
}

// ---------------------------------------------------------------------------
// Kernel 3: exact-adjoint backprojection (gather, deterministic) + update.
// Stages res[b] (128 KB) into LDS via CDNA5 async-to-LDS, then each thread
// owns one pixel and loops views with the exact 2x2 candidate window
// (sample pitch dt = 1.4198 > sqrt(2)*hat-support/2 -> only {floor, floor+1}
//  in both detector and ray-sample index can carry nonzero weight).
// out = x - cw[b] * (A^T r)
// ---------------------------------------------------------------------------
__global__ void bwd_update_kernel(const float* __restrict__ img,
                                  const float* __restrict__ res,
                                  const float* __restrict__ cw,
                                  float* __restrict__ out) {
  __shared__ float smem[VV * DD + 2 * VV];  // 32768 res + 128 ct + 128 st
  float* __restrict__ res_s = smem;
  float* __restrict__ ct_s  = smem + VV * DD;
  float* __restrict__ st_s  = ct_s + VV;

  const int b   = blockIdx.x >> 8;        // grid = B*256
  const int pb  = blockIdx.x & 255;
  const int tid = threadIdx.x;            // 256 threads

  // --- async stage res[b] (8192 x b128) into LDS ---
  const float* __restrict__ resb = res + (size_t)b * VV * DD;
  const unsigned lds_base = (unsigned)(uintptr_t)(void*)res_s;  // low 32b = LDS off
  #pragma unroll 4
  for (int it = 0; it < (VV * DD / 4) / 256; ++it) {   // 32 iterations
    const int f4 = it * 256 + tid;
    const unsigned loff = lds_base + (unsigned)(f4 * 16);
    const unsigned goff = (unsigned)(f4 * 16);
    asm volatile("global_load_async_to_lds_b128 %0, %1, %2"
                 :: "v"(loff), "v"(goff), "s"(resb) : "memory");
  }
  if (tid < VV) {
    const float th = (float)((double)tid * (3.14159265358979323846 / (double)VV));
    float st, ct;
    sincosf(th, &st, &ct);
    ct_s[tid] = ct;
    st_s[tid] = st;
  }
  asm volatile("s_wait_asynccnt 0x0" ::: "memory");
  __syncthreads();

  // --- pixel-driven exact adjoint ---
  const int p = pb * 256 + tid;           // 0 .. 65535
  const int i = p >> 8;
  const int j = p & 255;
  const float dx = (float)j - GEO_CC;
  const float dy = (float)i - GEO_CC;
  const float inv_dt = 1.0f / GEO_DT;

  float acc = 0.0f;
  for (int v = 0; v < VV; ++v) {
    const float ct = ct_s[v];
    const float st = st_s[v];
    const float tp = dx * ct + dy * st;        // detector coord of pixel
    const float sp = dy * ct - dx * st;        // ray-sample coord of pixel
    const float fd = (tp + GEO_L) * inv_dt;    // in (0.5, 254.5) for all pixels
    const float fk = (sp + GEO_L) * inv_dt;
    const int m  = (int)floorf(fd);
    const int mk = (int)floorf(fk);
    #pragma unroll
    for (int dd = 0; dd < 2; ++dd) {
      int di = m + dd;  di = min(max(di, 0), DD - 1);
      const float t = fmaf((float)di, GEO_DT, -GEO_L);
      float wpair = 0.0f;
      #pragma unroll
      for (int kk = 0; kk < 2; ++kk) {
        int ki = mk + kk;  ki = min(max(ki, 0), SS - 1);
        const float s  = fmaf((float)ki, GEO_DT, -GEO_L);
        const float x  = t * ct - s * st + GEO_CC;   // identical fwd expressions
        const float y  = t * st + s * ct + GEO_CC;
        const float hu = 1.0f - fabsf(x - (float)j);
        const float hw = 1.0f - fabsf(y - (float)i);
        wpair += fmaxf(hu, 0.0f) * fmaxf(hw, 0.0f);
      }
      acc = fmaf(wpair, res_s[v * DD + di], acc);
    }
  }

  const size_t gp = (size_t)b * NN * NN + (size_t)p;
  out[gp] = img[gp] - cw[b] * (acc * GEO_DT);
}

// ---------------------------------------------------------------------------
extern "C" void kernel_launch(void* const* d_in, const int* in_sizes, int n_in,
                              void* d_out, int out_size, void* d_ws, size_t ws_size,
                              hipStream_t stream) {
  (void)in_sizes; (void)n_in; (void)out_size; (void)ws_size;
  const float* img   = (const float*)d_in[0];  // [B,1,256,256]
  const float* proj  = (const float*)d_in[1];  // [B,1,128,256]
  const float* wf    = (const float*)d_in[2];  // [1]
  const float* gamma = (const float*)d_in[3];  // [1]
  const float* beta  = (const float*)d_in[4];  // [1]
  const float* fv    = (const float*)d_in[5];  // [B,7]
  const float* wv    = (const float*)d_in[6];  // [7,1]
  const float* bvec  = (const float*)d_in[7];  // [1]
  float* out = (float*)d_out;

  float* res = (float*)d_ws;                   // B*V*D floats (256 KB)
  float* cw  = res + (size_t)BB * VV * DD;     // B floats

  cw_kernel<<<1, 32, 0, stream>>>(wf, gamma, beta, fv, wv, bvec, cw);
  fwd_res_kernel<<<BB * VV, 1024, 0, stream>>>(img, proj, res);
  bwd_update_kernel<<<BB * 256, 256, 0, stream>>>(img, res, cw, out);
}